// GraphCNN_65472481460593
// MI455X (gfx1250) — compile-verified
//
#include <hip/hip_runtime.h>
#include <hip/hip_bf16.h>

#define GN 100000
#define GE 1200000
#define GG 1000
#define BN_EPS 1e-5f
#define LDP 80   // LDS row pitch in floats (80 mod 64 = 16 -> conflict-free halves)

typedef float v2f __attribute__((ext_vector_type(2)));
typedef float v8f __attribute__((ext_vector_type(8)));

// ---------------------------------------------------------------- utilities
__global__ void fill_kernel(float* __restrict__ p, float val, int n) {
    int i = blockIdx.x * blockDim.x + threadIdx.x;
    if (i < n) p[i] = val;
}

__global__ void deg_accum_kernel(const int* __restrict__ dst, float* __restrict__ deg, int e) {
    int i = blockIdx.x * blockDim.x + threadIdx.x;
    if (i < e) atomicAdd(&deg[dst[i]], 1.0f);
}

__global__ void deg_final_kernel(const float* __restrict__ deg, float* __restrict__ dis,
                                 float* __restrict__ dinv, int n) {
    int i = blockIdx.x * blockDim.x + threadIdx.x;
    if (i < n) { float d = deg[i]; dis[i] = rsqrtf(d); dinv[i] = 1.0f / d; }
}

__global__ void norm_kernel(const int* __restrict__ src, const int* __restrict__ dst,
                            const float* __restrict__ dis, float* __restrict__ nrm, int e) {
    int i = blockIdx.x * blockDim.x + threadIdx.x;
    if (i < e) nrm[i] = dis[src[i]] * dis[dst[i]];
}

// ---------------- async global -> LDS copy (gfx1250, tracked by ASYNCcnt)
__device__ __forceinline__ void async_g2l_b128(unsigned lds_off, const float* gptr) {
    asm volatile("global_load_async_to_lds_b128 %0, %1, off"
                 :: "v"(lds_off), "v"((unsigned long long)(uintptr_t)gptr)
                 : "memory");
}
__device__ __forceinline__ void wait_async0() {
    asm volatile("s_wait_asynccnt 0" ::: "memory");
}

// ------------------------------------------------- WMMA fp32 GEMM: Y = f(H) @ W
// H: [nrows,64] row-major, W: [64,64] row-major, Y: [nrows,64]
// f = identity (fuse_bn==0) or relu(h*scale[k]+shift[k]) applied per input feature.
// 128 threads = 4 waves; each wave owns a 16-row strip of the 64-row block.
// Staging uses GLOBAL_LOAD_ASYNC_TO_LDS_B128 (ASYNCcnt); tail rows are clamped,
// so garbage only enters dead output rows which are never stored.
__global__ void __launch_bounds__(128) gemm64_wmma_kernel(const float* __restrict__ H,
                                                          const float* __restrict__ W,
                                                          const float* __restrict__ scale,
                                                          const float* __restrict__ shift,
                                                          float* __restrict__ Y,
                                                          int nrows, int fuse_bn) {
    __shared__ float Wl[64 * LDP];
    __shared__ float Al[64 * LDP];
    const int t = threadIdx.x;
    const int base = blockIdx.x * 64;

    // async-stage W (64x64) and the 64-row activation tile into LDS
    #pragma unroll
    for (int i = 0; i < 8; ++i) {
        int idx = t + i * 128;            // float4 index, 0..1023
        int row = idx >> 4;
        int c4  = (idx & 15) << 2;
        async_g2l_b128((unsigned)(uintptr_t)&Wl[row * LDP + c4], W + row * 64 + c4);
        int gr = base + row;
        if (gr > nrows - 1) gr = nrows - 1;   // clamp: dead rows get garbage, never stored
        async_g2l_b128((unsigned)(uintptr_t)&Al[row * LDP + c4], H + (size_t)gr * 64 + c4);
    }
    wait_async0();
    __syncthreads();

    const int wave = t >> 5;
    const int lane = t & 31;
    const int m  = lane & 15;       // M for A, N for B, column for C/D
    const int hi = lane >> 4;
    const int kh = hi << 1;         // K sub-offset for A/B fragments

    // preload all 16 A fragments; optionally apply previous layer's BN+ReLU here
    v2f afrag[16];
    #pragma unroll
    for (int kt = 0; kt < 16; ++kt) {
        const float* p = &Al[(wave * 16 + m) * LDP + (kt << 2) + kh];
        v2f a; a.x = p[0]; a.y = p[1];
        afrag[kt] = a;
    }
    if (fuse_bn) {                  // uniform branch: EXEC stays all-ones
        #pragma unroll
        for (int kt = 0; kt < 16; ++kt) {
            int k0 = (kt << 2) + kh;
            v2f a = afrag[kt];
            a.x = fmaxf(a.x * scale[k0]     + shift[k0],     0.0f);
            a.y = fmaxf(a.y * scale[k0 + 1] + shift[k0 + 1], 0.0f);
            afrag[kt] = a;
        }
    }

    #pragma unroll
    for (int nt = 0; nt < 4; ++nt) {
        v8f acc = {};
        #pragma unroll
        for (int kt = 0; kt < 16; ++kt) {
            int k0 = (kt << 2) + kh;
            v2f b;
            b.x = Wl[k0 * LDP + nt * 16 + m];
            b.y = Wl[(k0 + 1) * LDP + nt * 16 + m];
            acc = __builtin_amdgcn_wmma_f32_16x16x4_f32(
                false, afrag[kt], false, b, (short)0, acc, false, false);
        }
        // scatter C/D: VGPR r -> row (r + 8*hi), column (nt*16 + m)
        #pragma unroll
        for (int r = 0; r < 8; ++r) {
            int grow = base + wave * 16 + r + (hi << 3);
            if (grow < nrows) Y[grow * 64 + nt * 16 + m] = acc[r];
        }
    }
}

// --------------------------------------------- GCN aggregation (self + edges)
__global__ void agg_init_kernel(const float* __restrict__ xw, const float* __restrict__ dinv,
                                const float* __restrict__ bias, float* __restrict__ agg, int n) {
    int i = blockIdx.x * blockDim.x + threadIdx.x;
    if (i < n * 64) {
        int node = i >> 6, f = i & 63;
        agg[i] = xw[i] * dinv[node] + bias[f];
    }
}

__global__ void __launch_bounds__(256) edge_scatter_kernel(const float* __restrict__ xw,
                                                           const int* __restrict__ src,
                                                           const int* __restrict__ dst,
                                                           const float* __restrict__ nrm,
                                                           float* __restrict__ agg, int e) {
    int t  = threadIdx.x;
    int ei = t >> 6, f = t & 63;
    for (int eb = blockIdx.x * 4; eb < e; eb += gridDim.x * 4) {
        int ee = eb + ei;
        if (ee < e) {
            int   s  = src[ee];
            int   d  = dst[ee];
            float nv = nrm[ee];
            atomicAdd(&agg[d * 64 + f], xw[s * 64 + f] * nv);
        }
    }
}

// ------------------------------------------------------------ BatchNorm(eval)
__global__ void __launch_bounds__(256) bn_stats_kernel(const float* __restrict__ h,
                                                       float* __restrict__ bsum,
                                                       float* __restrict__ bsq, int nrows) {
    __shared__ float ls[256];
    __shared__ float ls2[256];
    int t = threadIdx.x;
    int f = t & 63;
    float s = 0.f, s2 = 0.f;
    for (int r = blockIdx.x * 4 + (t >> 6); r < nrows; r += gridDim.x * 4) {
        float v = h[r * 64 + f];
        s += v; s2 += v * v;
    }
    ls[t] = s; ls2[t] = s2;
    __syncthreads();
    if (t < 64) {
        s  = ls[t]  + ls[t + 64]  + ls[t + 128]  + ls[t + 192];
        s2 = ls2[t] + ls2[t + 64] + ls2[t + 128] + ls2[t + 192];
        atomicAdd(&bsum[f], s);
        atomicAdd(&bsq[f], s2);
    }
}

__global__ void bn_final_kernel(const float* __restrict__ bsum, const float* __restrict__ bsq,
                                const float* __restrict__ gam, const float* __restrict__ bet,
                                float* __restrict__ scale, float* __restrict__ shift) {
    int f = threadIdx.x;  // 64 threads
    const float invN = 1.0f / (float)GN;
    float mean = bsum[f] * invN;
    float var  = bsq[f] * invN - mean * mean;
    float sc   = gam[f] * rsqrtf(var + BN_EPS);
    scale[f] = sc;
    shift[f] = bet[f] - mean * sc;
}

__global__ void bn_apply_kernel(const float* __restrict__ agg, const float* __restrict__ scale,
                                const float* __restrict__ shift, float* __restrict__ hout, int n) {
    int i = blockIdx.x * blockDim.x + threadIdx.x;
    if (i < n * 64) {
        int f = i & 63;
        hout[i] = fmaxf(agg[i] * scale[f] + shift[f], 0.0f);
    }
}

// ------------------------------------------------------------------- pooling
__global__ void pool_accum_kernel(const float* __restrict__ h, const int* __restrict__ batch,
                                  float* __restrict__ pooled, float* __restrict__ cnt, int n) {
    int i = blockIdx.x * blockDim.x + threadIdx.x;
    if (i < n * 64) {
        int node = i >> 6, f = i & 63;
        int g = batch[node];
        atomicAdd(&pooled[g * 64 + f], h[i]);
        if (f == 0) atomicAdd(&cnt[g], 1.0f);
    }
}

// --------------------------------------------------- MLP head + log_softmax
__global__ void __launch_bounds__(32) head_kernel(const float* __restrict__ pooled,
                                                  const float* __restrict__ cnt,
                                                  const float* __restrict__ fc1w,
                                                  const float* __restrict__ fc1b,
                                                  const float* __restrict__ fc2w,
                                                  const float* __restrict__ fc2b,
                                                  float* __restrict__ out) {
    __shared__ float p[64];
    __shared__ float z1[32];
    __shared__ float z2[16];
    int g = blockIdx.x;
    int lane = threadIdx.x;
    float inv = 1.0f / fmaxf(cnt[g], 1.0f);
    p[lane]      = pooled[g * 64 + lane] * inv;
    p[lane + 32] = pooled[g * 64 + 32 + lane] * inv;
    __syncthreads();
    float a = fc1b[lane];
    for (int k = 0; k < 64; ++k) a += p[k] * fc1w[k * 32 + lane];
    z1[lane] = fmaxf(a, 0.0f);
    __syncthreads();
    if (lane < 10) {
        float z = fc2b[lane];
        for (int k = 0; k < 32; ++k) z += z1[k] * fc2w[k * 10 + lane];
        z2[lane] = z;
    }
    __syncthreads();
    if (lane < 10) {
        float mx = z2[0];
        for (int i = 1; i < 10; ++i) mx = fmaxf(mx, z2[i]);
        float se = 0.f;
        for (int i = 0; i < 10; ++i) se += expf(z2[i] - mx);
        out[g * 10 + lane] = z2[lane] - mx - logf(se);
    }
}

// ------------------------------------------------------------------- driver
extern "C" void kernel_launch(void* const* d_in, const int* in_sizes, int n_in,
                              void* d_out, int out_size, void* d_ws, size_t ws_size,
                              hipStream_t stream) {
    const float* x     = (const float*)d_in[0];
    const int*   ei    = (const int*)d_in[1];
    const int*   batch = (const int*)d_in[2];
    const float* W[3]  = {(const float*)d_in[3],  (const float*)d_in[7],  (const float*)d_in[11]};
    const float* bb[3] = {(const float*)d_in[4],  (const float*)d_in[8],  (const float*)d_in[12]};
    const float* gm[3] = {(const float*)d_in[5],  (const float*)d_in[9],  (const float*)d_in[13]};
    const float* be[3] = {(const float*)d_in[6],  (const float*)d_in[10], (const float*)d_in[14]};
    const float* fc1w  = (const float*)d_in[15];
    const float* fc1b  = (const float*)d_in[16];
    const float* fc2w  = (const float*)d_in[17];
    const float* fc2b  = (const float*)d_in[18];
    float* out = (float*)d_out;
    const int* src = ei;
    const int* dst = ei + GE;

    // workspace carve-up (floats)
    float* w = (float*)d_ws;
    float* deg    = w; w += GN;
    float* dis    = w; w += GN;
    float* dinv   = w; w += GN;
    float* nrm    = w; w += GE;
    float* bufP   = w; w += (size_t)GN * 64;
    float* bufQ   = w; w += (size_t)GN * 64;
    float* bsum   = w; w += 64;   // bsq contiguous after bsum
    float* bsq    = w; w += 64;
    float* bscale = w; w += 64;
    float* bshift = w; w += 64;
    float* pooled = w; w += GG * 64;  // cnt contiguous after pooled
    float* cnt    = w; w += GG;

    // degree with self-loop, deg^{-1/2}, deg^{-1}, per-edge norm (computed once)
    fill_kernel<<<(GN + 255) / 256, 256, 0, stream>>>(deg, 1.0f, GN);
    deg_accum_kernel<<<(GE + 255) / 256, 256, 0, stream>>>(dst, deg, GE);
    deg_final_kernel<<<(GN + 255) / 256, 256, 0, stream>>>(deg, dis, dinv, GN);
    norm_kernel<<<(GE + 255) / 256, 256, 0, stream>>>(src, dst, dis, nrm, GE);

    // 2-buffer schedule:
    //   layer l: gemm(in -> P) [BN(l-1)+ReLU fused into input for l>0]
    //            agg_init(P -> Q), scatter(P -> Q), stats/final(Q) -> scale/shift
    //   in(l=0)=x, in(l>0)=Q (raw aggregates of previous layer)
    const float* hin = x;
    for (int l = 0; l < 3; ++l) {
        gemm64_wmma_kernel<<<(GN + 63) / 64, 128, 0, stream>>>(
            hin, W[l], bscale, bshift, bufP, GN, l > 0 ? 1 : 0);
        agg_init_kernel<<<(GN * 64 + 255) / 256, 256, 0, stream>>>(bufP, dinv, bb[l], bufQ, GN);
        edge_scatter_kernel<<<30000, 256, 0, stream>>>(bufP, src, dst, nrm, bufQ, GE);
        fill_kernel<<<1, 128, 0, stream>>>(bsum, 0.0f, 128);  // zero bsum + bsq
        bn_stats_kernel<<<512, 256, 0, stream>>>(bufQ, bsum, bsq, GN);
        bn_final_kernel<<<1, 64, 0, stream>>>(bsum, bsq, gm[l], be[l], bscale, bshift);
        hin = bufQ;
    }
    // materialize final activated features for pooling
    bn_apply_kernel<<<(GN * 64 + 255) / 256, 256, 0, stream>>>(bufQ, bscale, bshift, bufP, GN);

    fill_kernel<<<(GG * 64 + GG + 255) / 256, 256, 0, stream>>>(pooled, 0.0f, GG * 64 + GG);
    pool_accum_kernel<<<(GN * 64 + 255) / 256, 256, 0, stream>>>(bufP, batch, pooled, cnt, GN);
    head_kernel<<<GG, 32, 0, stream>>>(pooled, cnt, fc1w, fc1b, fc2w, fc2b, out);
}